// NonMaxSuppression_71451075936890
// MI455X (gfx1250) — compile-verified
//
#include <hip/hip_runtime.h>
#include <stdint.h>

// -----------------------------------------------------------------------------
// Greedy NMS, N=8192, MI455X (gfx1250).
// Latency-bound sequential problem: everything lives in one WGP's 320KB LDS.
// CDNA5 paths used:
//   * tensor_load_to_lds (TDM, D# built per ISA ch.8) + s_wait_tensorcnt:
//     one-instruction strided DMA of the confidence column into LDS.
//   * global_load_async_to_lds_b32 + s_wait_asynccnt (ASYNCcnt): direct
//     global->LDS gather of sorted box rows, no VGPR round trip.
//   * global_prefetch_b8 (__builtin_prefetch) to warm caches for the gather.
//   * wave32 __shfl/__ballot chunk solver for the sequential greedy core.
// -----------------------------------------------------------------------------

#ifndef __has_builtin
#define __has_builtin(x) 0
#endif
#if __has_builtin(__builtin_amdgcn_tensor_load_to_lds)
#define HAVE_TDM 1
#else
#define HAVE_TDM 0
#endif

typedef unsigned int u32x4 __attribute__((ext_vector_type(4)));
typedef int          i32x4 __attribute__((ext_vector_type(4)));
typedef int          i32x8 __attribute__((ext_vector_type(8)));

#define THREADS 1024
#define NMAX    8192

// Dynamic-LDS layout (byte offsets; dynamic LDS base == 0, confirmed in rd-1 asm)
static constexpr uint32_t OFF_KEYS  = 0;        // uint64[8192]  65536 B (sort keys; low32 = orig idx)
static constexpr uint32_t OFF_BOX   = 65536;    // float4[8192] 131072 B (sorted x1,y1,x2,y2)
static constexpr uint32_t OFF_AREA  = 196608;   // float [8192]  32768 B (areas; conf staging early)
static constexpr uint32_t OFF_FLAG  = 229376;   // uint8 [8192]   8192 B (keep flags)
static constexpr uint32_t OFF_CBOX  = 237568;   // float4[32]      512 B (chunk survivor boxes)
static constexpr uint32_t OFF_CAREA = 238080;   // float [32]      128 B
static constexpr uint32_t OFF_MISC  = 238208;   // int   [4]        16 B (chunk survivor count)
static constexpr uint32_t OFF_SCAN  = 238224;   // int   [1024]   4096 B (prefix-sum scratch)
static constexpr uint32_t LDS_BYTES = 242320;   // ~236.6 KB < 320 KB/WGP

__device__ __forceinline__ float iou_f(float4 a, float aa, float4 b, float ab) {
  float ix1 = fmaxf(a.x, b.x);
  float iy1 = fmaxf(a.y, b.y);
  float ix2 = fminf(a.z, b.z);
  float iy2 = fminf(a.w, b.w);
  float iw  = fmaxf(ix2 - ix1, 0.0f);
  float ih  = fmaxf(iy2 - iy1, 0.0f);
  float inter = iw * ih;
  return inter / (aa + ab - inter + 1e-12f);   // matches reference epsilon
}

__global__ __launch_bounds__(THREADS)
void nms_fused_kernel(const float* __restrict__ preds,
                      const float* __restrict__ pTh,
                      const int*   __restrict__ pMaxP,
                      float*       __restrict__ out,
                      int N, int NP) {
  extern __shared__ char smem[];
  unsigned long long* keys = (unsigned long long*)(smem + OFF_KEYS);
  float4*  boxes = (float4*)(smem + OFF_BOX);
  float*   area  = (float*) (smem + OFF_AREA);
  uint8_t* flags = (uint8_t*)(smem + OFF_FLAG);
  float4*  cbox  = (float4*)(smem + OFF_CBOX);
  float*   carea = (float*) (smem + OFF_CAREA);
  int*     misc  = (int*)   (smem + OFF_MISC);
  int*     scan  = (int*)   (smem + OFF_SCAN);

  const int   tid  = threadIdx.x;
  const float th   = *pTh;
  const int   maxp = *pMaxP;

  // ---- Stage 0: stage the confidence column (preds[:,4]) into LDS.
#if HAVE_TDM
  // Tensor Data Mover: one strided DMA. D# per CDNA5 ISA ch.8:
  //   2D tensor, data_size=4B, tile 1 x N, tensor_dim0_stride=5 elements,
  //   global_addr = &preds[4], lds_addr = OFF_AREA (contiguous dest).
  if (tid == 0) {
    uint64_t ga = (uint64_t)(uintptr_t)(preds + 4);
    u32x4 g0 = {
      1u,                                                  // count=1, user mode
      OFF_AREA,                                            // lds_addr
      (uint32_t)ga,                                        // global_addr[31:0]
      (uint32_t)((ga >> 32) & 0x01FFFFFFu) | 0x80000000u   // addr[56:32] | type=2
    };
    uint32_t n = (uint32_t)N;
    i32x8 g1 = {
      (int)(2u << 16),                                     // data_size=4B; no mask/pad/iter
      (int)(1u << 16),                                     // tensor_dim0 = 1 (low 16)
      (int)((n & 0xFFFFu) << 16),                          // tensor_dim1[15:0] = N
      (int)(((n >> 16) & 0xFFFFu) | (1u << 16)),           // tensor_dim1[31:16] | tile_dim0=1
      (int)(n & 0xFFFFu),                                  // tile_dim1 = N; tile_dim2 = 0
      5,                                                   // tensor_dim0_stride = 5 elems
      0, 0                                                 // strides hi / dim1_stride = 0
    };
    i32x4 gz = { 0, 0, 0, 0 };
#if __clang_major__ >= 23
    i32x8 gz8 = { 0, 0, 0, 0, 0, 0, 0, 0 };
    __builtin_amdgcn_tensor_load_to_lds(g0, g1, gz, gz, gz8, 0);
#else
    __builtin_amdgcn_tensor_load_to_lds(g0, g1, gz, gz, 0);
#endif
#if __has_builtin(__builtin_amdgcn_s_wait_tensorcnt)
    __builtin_amdgcn_s_wait_tensorcnt(0);
#else
    asm volatile("s_wait_tensorcnt 0" ::: "memory");
#endif
  }
  // Warm caches for the post-sort gather of box rows.
  for (int p = tid; p < N; p += THREADS)
    __builtin_prefetch(preds + (size_t)p * 5, 0, 3);
#else
  // Fallback: per-lane async global->LDS copy of the conf column.
  for (int p = tid; p < N; p += THREADS) {
    uint32_t ldsoff = (uint32_t)(uintptr_t)(area + p);
    uint64_t ga     = (uint64_t)(uintptr_t)(preds + (size_t)p * 5 + 4);
    asm volatile("global_load_async_to_lds_b32 %0, %1, off"
                 :: "v"(ldsoff), "v"(ga) : "memory");
    __builtin_prefetch(preds + (size_t)p * 5, 0, 3);
  }
  asm volatile("s_wait_asynccnt 0" ::: "memory");
#endif
  __syncthreads();

  // ---- Stage 1: build 64-bit sort keys.
  // key = (~orderable(conf)) << 32 | idx  -> ascending u64 sort gives
  // stable descending-conf order (== jnp.argsort(-conf)).
  for (int p = tid; p < NP; p += THREADS) {
    unsigned long long key;
    if (p < N) {
      uint32_t u = __float_as_uint(area[p]);                 // staged conf
      u = (u & 0x80000000u) ? ~u : (u | 0x80000000u);        // ascending-orderable
      key = ((unsigned long long)(~u) << 32) | (uint32_t)p;  // descending + stable tie
    } else {
      key = 0xFFFFFFFFFFFFFFFFull;                           // padding sorts last
    }
    keys[p] = key;
  }

  // ---- Stage 2: bitonic sort of u64 keys in LDS.
  for (unsigned k = 2; k <= (unsigned)NP; k <<= 1) {
    for (unsigned j = k >> 1; j > 0; j >>= 1) {
      __syncthreads();
      for (unsigned e = tid; e < (unsigned)NP; e += THREADS) {
        unsigned partner = e ^ j;
        if (partner > e) {
          bool up = ((e & k) == 0);
          unsigned long long A = keys[e], B = keys[partner];
          if ((A > B) == up) { keys[e] = B; keys[partner] = A; }
        }
      }
    }
  }
  __syncthreads();

  // ---- Stage 3a: gather sorted box rows straight into LDS (async, no VGPR
  //      round trip). 4x b32 keeps natural alignment on the 20-byte row stride.
  for (int p = tid; p < NP; p += THREADS) {
    uint32_t oi    = (uint32_t)keys[p];
    uint32_t lbase = (uint32_t)(uintptr_t)(&boxes[p]);
    if (oi < (uint32_t)N) {
      uint64_t gb = (uint64_t)(uintptr_t)(preds + (size_t)oi * 5);
      #pragma unroll
      for (int q = 0; q < 4; ++q) {
        uint32_t lo = lbase + (uint32_t)q * 4u;
        uint64_t ga = gb + (uint64_t)q * 4u;
        asm volatile("global_load_async_to_lds_b32 %0, %1, off"
                     :: "v"(lo), "v"(ga) : "memory");
      }
    } else {
      boxes[p] = make_float4(0.f, 0.f, 0.f, 0.f);
    }
  }
  asm volatile("s_wait_asynccnt 0" ::: "memory");
  __syncthreads();

  // ---- Stage 3b: areas + keep flags from the LDS-resident boxes.
  for (int p = tid; p < NP; p += THREADS) {
    float4 b = boxes[p];
    area[p]  = (b.z - b.x) * (b.w - b.y);
    flags[p] = ((uint32_t)keys[p] < (uint32_t)N) ? 1 : 0;
  }

  // ---- Stage 4: chunked greedy suppression.
  // Per 32-wide chunk: wave 0 resolves the intra-chunk triangle with wave32
  // shuffles (no barriers), publishes the survivor list; then all 1024 threads
  // suppress the tail j > chunk in parallel, with early-out on dead boxes.
  const int nCh = NP >> 5;
  for (int c = 0; c < nCh; ++c) {
    __syncthreads();                       // tail updates of chunk c-1 visible
    if (tid < 32) {
      const int l = tid, g = (c << 5) + l;
      unsigned fu = flags[g];
      float4 b = boxes[g];
      float  a = area[g];
      for (int i = 0; i < 31; ++i) {
        unsigned fi = __shfl(fu, i);       // lane i's *current* keep state
        float bx = __shfl(b.x, i), by = __shfl(b.y, i);
        float bz = __shfl(b.z, i), bw = __shfl(b.w, i);
        float ai = __shfl(a,   i);
        if (fi && l > i) {
          float4 bi = make_float4(bx, by, bz, bw);
          if (iou_f(bi, ai, b, a) > th) fu = 0;
        }
      }
      flags[g] = (uint8_t)(fu ? 1 : 0);
      unsigned bal = (unsigned)__ballot(fu != 0);
      if (l == 0) misc[0] = __popc(bal);
      if (fu) {
        int pos = __popc(bal & ((1u << l) - 1u));
        cbox[pos]  = b;
        carea[pos] = a;
      }
    }
    __syncthreads();
    const int cc = misc[0];
    if (cc > 0) {
      for (int jj = ((c + 1) << 5) + tid; jj < NP; jj += THREADS) {
        if (!flags[jj]) continue;          // critical early-out
        float4 bj = boxes[jj];
        float  aj = area[jj];
        for (int s = 0; s < cc; ++s) {
          if (iou_f(cbox[s], carea[s], bj, aj) > th) { flags[jj] = 0; break; }
        }
      }
    }
  }

  // ---- Stage 5: compaction (block prefix sum over keep flags) + output.
  __syncthreads();
  const int per = (NP + THREADS - 1) / THREADS;
  const int p0  = tid * per;
  int cnt = 0;
  for (int k = 0; k < per; ++k) {
    int p = p0 + k;
    if (p < NP && flags[p]) cnt++;
  }
  scan[tid] = cnt;
  __syncthreads();
  for (int off = 1; off < THREADS; off <<= 1) {
    int v = scan[tid];
    int w = (tid >= off) ? scan[tid - off] : 0;
    __syncthreads();
    scan[tid] = v + w;
    __syncthreads();
  }
  const int total = scan[THREADS - 1];
  int r = scan[tid] - cnt;                 // exclusive prefix for my segment
  for (int k = 0; k < per; ++k) {
    int p = p0 + k;
    if (p < NP && flags[p]) {
      if (r < maxp) {
        int oi = (int)(uint32_t)keys[p];   // original index of p-th sorted box
        out[(size_t)maxp * 5 + r] = (float)oi;
        #pragma unroll
        for (int q = 0; q < 5; ++q)
          out[(size_t)r * 5 + q] = preds[(size_t)oi * 5 + q];
      }
      r++;
    }
  }
  for (int m = total + tid; m < maxp; m += THREADS) {
    out[(size_t)maxp * 5 + m] = -1.0f;     // keep_idx = -1 for invalid slots
    #pragma unroll
    for (int q = 0; q < 5; ++q) out[(size_t)m * 5 + q] = 0.0f;
  }
}

extern "C" void kernel_launch(void* const* d_in, const int* in_sizes, int n_in,
                              void* d_out, int out_size, void* d_ws, size_t ws_size,
                              hipStream_t stream) {
  (void)n_in; (void)out_size; (void)d_ws; (void)ws_size;
  const float* preds = (const float*)d_in[0];
  const float* pTh   = (const float*)d_in[1];
  const int*   pMaxP = (const int*)  d_in[2];
  float* out = (float*)d_out;

  int N = in_sizes[0] / 5;
  if (N > NMAX) N = NMAX;
  int NP = 32;
  while (NP < N) NP <<= 1;                 // pow2 for bitonic sort (8192 here)

  hipFuncSetAttribute(reinterpret_cast<const void*>(nms_fused_kernel),
                      hipFuncAttributeMaxDynamicSharedMemorySize,
                      (int)LDS_BYTES);
  nms_fused_kernel<<<dim3(1), dim3(THREADS), LDS_BYTES, stream>>>(
      preds, pTh, pMaxP, out, N, NP);
}